// PCA_83425444758100
// MI455X (gfx1250) — compile-verified
//
#include <hip/hip_runtime.h>
#include <hip/hip_bf16.h>

#define NHEADS 16
#define BB 8
#define SS 4096
#define DD 1024
#define HS 64
#define BH (BB * NHEADS) /* 128 */
#define LN_EPS 1e-3f

#define SCHUNKS 8
#define ROWS_PER_WG (SS / SCHUNKS)        /* 512 */
#define WAVES 8
#define ROWS_PER_WAVE (ROWS_PER_WG / WAVES) /* 64 */
#define NSWEEPS 12

typedef __attribute__((ext_vector_type(2))) float v2f;
typedef __attribute__((ext_vector_type(8))) float v8f;

__global__ void zero_ws_kernel(float* ws, int n) {
    int i = blockIdx.x * blockDim.x + threadIdx.x;
    if (i < n) ws[i] = 0.0f;
}

// ---------------------------------------------------------------------------
// Kernel 1: per-(b,h) Gram = sum_s x_s x_s^T via V_WMMA_F32_16X16X4_F32,
// plus per-column sums (for the centering rank-1 correction).
// grid = (BH, SCHUNKS), block = 256 (8 waves). Each wave handles 64 rows of S.
// A (16x4) and B (4x16) operands share identical lane layouts -> one v2f load
// per 16-column group feeds both sides; only upper-triangle tiles computed.
// ---------------------------------------------------------------------------
__global__ __launch_bounds__(256) void gram_kernel(const float* __restrict__ x,
                                                   float* __restrict__ gram,
                                                   float* __restrict__ colsum) {
    __shared__ float sg[HS * HS];
    __shared__ float sc[HS];

    const int tid = threadIdx.x;
    const int bh = blockIdx.x;
    const int b = bh / NHEADS;
    const int h = bh % NHEADS;

    for (int e = tid; e < HS * HS; e += 256) sg[e] = 0.0f;
    if (tid < HS) sc[tid] = 0.0f;
    __syncthreads();

    const int wave = tid >> 5;
    const int lane = tid & 31;
    const int m = lane & 15;          // column-within-16 / N index
    const int kr = (lane >> 4) * 2;   // K rows 0..1 (lanes 0-15) or 2..3 (lanes 16-31)

    const int srow0 = blockIdx.y * ROWS_PER_WG + wave * ROWS_PER_WAVE;
    const float* xb = x + ((size_t)b * SS) * DD + (size_t)h * HS;

    v8f acc[10];
#pragma unroll
    for (int t = 0; t < 10; ++t)
#pragma unroll
        for (int r = 0; r < 8; ++r) acc[t][r] = 0.0f;

    float cs[4] = {0.0f, 0.0f, 0.0f, 0.0f};

    for (int ks = 0; ks < ROWS_PER_WAVE / 4; ++ks) {
        const int s0 = srow0 + ks * 4;
        const float* r0 = xb + (size_t)(s0 + kr) * DD;
        const float* r1 = xb + (size_t)(s0 + kr + 1) * DD;
        v2f op[4];
#pragma unroll
        for (int t = 0; t < 4; ++t) {
            op[t].x = r0[t * 16 + m];
            op[t].y = r1[t * 16 + m];
            cs[t] += op[t].x + op[t].y;
        }
#pragma unroll
        for (int ti = 0; ti < 4; ++ti) {
#pragma unroll
            for (int tj = ti; tj < 4; ++tj) {
                const int idx = ti * 4 + tj - (ti * (ti + 1)) / 2;
                acc[idx] = __builtin_amdgcn_wmma_f32_16x16x4_f32(
                    false, op[ti], false, op[tj], (short)0, acc[idx], false, false);
            }
        }
    }

    // C layout: VGPR r, lanes 0-15 -> M=r, lanes 16-31 -> M=8+r; N = lane&15.
    const int row_off = (lane >> 4) * 8;
#pragma unroll
    for (int ti = 0; ti < 4; ++ti) {
#pragma unroll
        for (int tj = ti; tj < 4; ++tj) {
            const int idx = ti * 4 + tj - (ti * (ti + 1)) / 2;
#pragma unroll
            for (int r = 0; r < 8; ++r) {
                const int gi = ti * 16 + row_off + r;
                const int gj = tj * 16 + m;
                atomicAdd(&sg[gi * HS + gj], acc[idx][r]);
            }
        }
    }
#pragma unroll
    for (int t = 0; t < 4; ++t) atomicAdd(&sc[t * 16 + m], cs[t]);
    __syncthreads();

    // Mirror lower-triangle tiles from upper (per-chunk mirror == mirror of sum).
    for (int e = tid; e < HS * HS; e += 256) {
        const int i = e >> 6, j = e & 63;
        if ((i >> 4) > (j >> 4)) sg[e] = sg[j * HS + i];
    }
    __syncthreads();

    float* g = gram + (size_t)bh * HS * HS;
    for (int e = tid; e < HS * HS; e += 256) atomicAdd(&g[e], sg[e]);
    if (tid < HS) atomicAdd(&colsum[bh * HS + tid], sc[tid]);
}

// ---------------------------------------------------------------------------
// Kernel 2: center Gram (rank-1 correction), parallel two-sided Jacobi eigh
// (32 disjoint pairs/round, 63-round tournament, NSWEEPS sweeps), ascending
// sort, pca[i,j] = w_s[j] * V[j][sigma(i)], LayerNorm(eps=1e-3) * 1/8.
// grid = BH, block = 256.
// ---------------------------------------------------------------------------
__global__ __launch_bounds__(256) void eigh_out_kernel(const float* __restrict__ gram,
                                                       const float* __restrict__ colsum,
                                                       const float* __restrict__ gamma,
                                                       const float* __restrict__ beta,
                                                       float* __restrict__ out) {
    __shared__ float A[HS][HS + 1];
    __shared__ float V[HS][HS + 1];
    __shared__ float mean[HS];
    __shared__ float cr[32], sr[32];
    __shared__ int pp[32], qq[32];
    __shared__ int top[32], bot[32];
    __shared__ float w[HS];
    __shared__ int sidx[HS];

    const int tid = threadIdx.x;
    const int bh = blockIdx.x;
    const float* g = gram + (size_t)bh * HS * HS;

    if (tid < HS) mean[tid] = colsum[bh * HS + tid] * (1.0f / (float)SS);
    __syncthreads();

    for (int e = tid; e < HS * HS; e += 256) {
        const int i = e >> 6, j = e & 63;
        A[i][j] = g[e] - (float)SS * mean[i] * mean[j];
        V[i][j] = (i == j) ? 1.0f : 0.0f;
    }
    if (tid < 32) { top[tid] = 2 * tid; bot[tid] = 2 * tid + 1; }
    __syncthreads();

    for (int sweep = 0; sweep < NSWEEPS; ++sweep) {
        for (int round = 0; round < 63; ++round) {
            if (tid < 32) {
                int p = top[tid], q = bot[tid];
                if (p > q) { int t = p; p = q; q = t; }
                pp[tid] = p; qq[tid] = q;
                const float apq = A[p][q];
                float c = 1.0f, s = 0.0f;
                if (fabsf(apq) > 1e-20f) {
                    const float tau = (A[q][q] - A[p][p]) / (2.0f * apq);
                    const float t = (tau >= 0.0f ? 1.0f : -1.0f) /
                                    (fabsf(tau) + sqrtf(1.0f + tau * tau));
                    c = rsqrtf(1.0f + t * t);
                    s = t * c;
                }
                cr[tid] = c; sr[tid] = s;
            }
            __syncthreads();
            // Column rotations on A and V (right-multiply by J): disjoint columns.
            for (int it = tid; it < 4096; it += 256) {
                const int pair = (it >> 6) & 31;
                const int r = it & 63;
                const int p = pp[pair], q = qq[pair];
                const float c = cr[pair], s = sr[pair];
                if (it < 2048) {
                    const float ap = A[r][p], aq = A[r][q];
                    A[r][p] = c * ap - s * aq;
                    A[r][q] = s * ap + c * aq;
                } else {
                    const float vp = V[r][p], vq = V[r][q];
                    V[r][p] = c * vp - s * vq;
                    V[r][q] = s * vp + c * vq;
                }
            }
            __syncthreads();
            // Row rotations on A (left-multiply by J^T): disjoint rows.
            for (int it = tid; it < 2048; it += 256) {
                const int pair = it >> 6;
                const int cc = it & 63;
                const int p = pp[pair], q = qq[pair];
                const float c = cr[pair], s = sr[pair];
                const float ap = A[p][cc], aq = A[q][cc];
                A[p][cc] = c * ap - s * aq;
                A[q][cc] = s * ap + c * aq;
            }
            __syncthreads();
            // Round-robin tournament rotation (top[0] fixed).
            if (tid == 0) {
                const int t31 = top[31];
                const int b0 = bot[0];
                for (int i = 31; i >= 2; --i) top[i] = top[i - 1];
                top[1] = b0;
                for (int i = 0; i < 31; ++i) bot[i] = bot[i + 1];
                bot[31] = t31;
            }
            __syncthreads();
        }
    }

    if (tid < HS) w[tid] = A[tid][tid];
    __syncthreads();
    if (tid < HS) {
        const float wi = w[tid];
        int cnt = 0;
        for (int k = 0; k < HS; ++k) {
            const float wk = w[k];
            if (wk < wi || (wk == wi && k < tid)) ++cnt;
        }
        sidx[cnt] = tid;   // ranks are unique -> no conflicts
    }
    __syncthreads();

    if (tid < HS) {
        const int i = tid;
        const int si = sidx[i];
        float y[HS];
        float msum = 0.0f;
#pragma unroll
        for (int j = 0; j < HS; ++j) {
            const float val = w[sidx[j]] * V[j][si];
            y[j] = val;
            msum += val;
        }
        const float mn = msum * (1.0f / (float)HS);
        float vsum = 0.0f;
#pragma unroll
        for (int j = 0; j < HS; ++j) { const float d = y[j] - mn; vsum += d * d; }
        const float inv = rsqrtf(vsum * (1.0f / (float)HS) + LN_EPS);
        float* o = out + ((size_t)bh * HS + i) * HS;
#pragma unroll
        for (int j = 0; j < HS; ++j) {
            o[j] = ((y[j] - mn) * inv * gamma[j] + beta[j]) * 0.125f;
        }
    }
}

extern "C" void kernel_launch(void* const* d_in, const int* in_sizes, int n_in,
                              void* d_out, int out_size, void* d_ws, size_t ws_size,
                              hipStream_t stream) {
    const float* x     = (const float*)d_in[0];
    const float* gamma = (const float*)d_in[1];
    const float* beta  = (const float*)d_in[2];
    float* out = (float*)d_out;

    float* gram   = (float*)d_ws;                       // BH * 64 * 64 floats
    float* colsum = gram + (size_t)BH * HS * HS;        // BH * 64 floats

    const int nzero = BH * HS * HS + BH * HS;
    zero_ws_kernel<<<(nzero + 255) / 256, 256, 0, stream>>>((float*)d_ws, nzero);
    gram_kernel<<<dim3(BH, SCHUNKS), 256, 0, stream>>>(x, gram, colsum);
    eigh_out_kernel<<<BH, 256, 0, stream>>>(gram, colsum, gamma, beta, out);
}